// NanEmbedWeird_25228637896775
// MI455X (gfx1250) — compile-verified
//
#include <hip/hip_runtime.h>

typedef float v2f __attribute__((ext_vector_type(2)));
typedef float v8f __attribute__((ext_vector_type(8)));
typedef unsigned int u32x4 __attribute__((ext_vector_type(4)));
typedef int i32x4 __attribute__((ext_vector_type(4)));
typedef int i32x8 __attribute__((ext_vector_type(8)));

#define BS   32
#define INS  1024   // K (in_size)
#define OUTS 1024   // M (out_size)
#define FE   2048   // N (feats)

#define BM 128      // macro tile in O
#define BN 64       // macro tile in F
#define BK 32       // K block

#define WS_STRIDE 36    // W tile row stride (floats): 32 data + 4 pad (TDM pad codes 4/3)
#define XS2_STRIDE 160  // interleaved X tile row stride (words); 160 mod 64 == 32

#if __has_builtin(__builtin_amdgcn_tensor_load_to_lds) && __has_builtin(__builtin_amdgcn_s_wait_tensorcnt)
#define HAVE_TDM 1
#else
#define HAVE_TDM 0
#endif

__global__ __launch_bounds__(256)
void bgemm_wmma_f32(const float* __restrict__ x,
                    const float* __restrict__ w,
                    float* __restrict__ out) {
    __shared__ float Ws[BM * WS_STRIDE];         // 18432 B, written by TDM
    __shared__ float Xs2[(BK / 2) * XS2_STRIDE]; // 10240 B, K-pair interleaved

    const int b  = blockIdx.z;
    const int f0 = blockIdx.x * BN;
    const int o0 = blockIdx.y * BM;

    const int tid  = threadIdx.x;
    const int lane = tid & 31;
    const int wave = tid >> 5;
    const int wr   = wave >> 1;        // 0..3  (O direction)
    const int wc   = wave & 1;         // 0..1  (F direction)
    const int m_wave = wr * 32;
    const int n_wave = wc * 32;

    const float* Wb = w   + (size_t)b * OUTS * INS + (size_t)o0 * INS;
    const float* Xb = x   + (size_t)b * INS  * FE  + f0;
    float*       Ob = out + (size_t)b * OUTS * FE  + (size_t)o0 * FE + f0;

    // lane decode for WMMA operand layouts (ISA 7.12.2, 32-bit A 16x4 / B 4x16)
    const int al = lane & 15;          // A-row / B-col within 16
    const int ak = (lane >> 4) << 1;   // K sub-offset: 0 (lanes 0-15) or 2 (lanes 16-31)

    // X staging decode: 256 threads x (2 global float4 loads -> 2 interleaved b128 stores)
    const int xpr = tid >> 4;          // pair-row 0..15
    const int xn4 = (tid & 15) << 2;   // n 0..60 step 4

#if HAVE_TDM
    const unsigned lds_ws = (unsigned)(uintptr_t)&Ws[0];   // LDS aperture: low 32 bits = LDS offset
    // D# group 1 (constant): data_size=4B (2<<16), pad_enable (1<<20),
    // pad_interval=32 DW (4<<22), pad_amount=4 DW (3<<25);
    // tensor_dim0=32 (bits 79:48), tensor_dim1=128 (bits 111:80),
    // tile_dim0=32 (bits 127:112), tile_dim1=128 (bits 143:128),
    // tensor_dim0_stride=1024 (bits 207:160).
    const i32x8 g1 = { (int)0x07120000u,
                       (int)(32u << 16),    // tensor_dim0 lo16 in [31:16]
                       (int)(128u << 16),   // tensor_dim1 lo16 in [31:16]
                       (int)(32u << 16),    // tile_dim0 in [31:16]
                       128,                 // tile_dim1
                       1024,                // tensor_dim0_stride lo32
                       0, 0 };
    const i32x4 gz4 = { 0, 0, 0, 0 };
    const i32x8 gz8 = { 0, 0, 0, 0, 0, 0, 0, 0 };
#endif

    v8f c00 = {}; v8f c01 = {}; v8f c10 = {}; v8f c11 = {};

    for (int kb = 0; kb < INS; kb += BK) {
#if HAVE_TDM
        // ---- W tile via Tensor Data Mover (wave 0 issues; EXEC-independent DMA) ----
        if (wave == 0) {
            unsigned long long gaddr = (unsigned long long)(uintptr_t)(Wb + kb);
            u32x4 g0 = { 1u,                                   // count=1 (valid descriptor)
                         lds_ws,                               // lds_addr
                         (unsigned)gaddr,                      // global_addr[31:0]
                         ((unsigned)(gaddr >> 32) & 0x01FFFFFFu) | (2u << 30) }; // addr[56:32] | type=2
            __builtin_amdgcn_tensor_load_to_lds(g0, g1, gz4, gz4, gz8, 0);
        }
#else
        // ---- fallback: stage W tile via vmem + ds ----
        {
            const float* src = Wb + kb;
            #pragma unroll
            for (int i = 0; i < 4; ++i) {
                int idx = i * 256 + tid;
                int row = idx >> 3;
                int c4  = (idx & 7) << 2;
                float4 v = *(const float4*)(src + (size_t)row * INS + c4);
                *(float4*)(&Ws[row * WS_STRIDE + c4]) = v;
            }
        }
#endif

        // ---- stage X tile, K-pair interleaved: Xs2[k>>1][2n | (k&1)] ----
        {
            const float* s0 = Xb + (size_t)(kb + 2 * xpr) * FE + xn4;
            float4 a = *(const float4*)s0;
            float4 c = *(const float4*)(s0 + FE);
            float* dst = &Xs2[xpr * XS2_STRIDE + (xn4 << 1)];
            *(float4*)(dst + 0) = make_float4(a.x, c.x, a.y, c.y);
            *(float4*)(dst + 4) = make_float4(a.z, c.z, a.w, c.w);
        }

#if HAVE_TDM
        if (wave == 0) {
            __builtin_amdgcn_s_wait_tensorcnt(0);
        }
#endif
        __syncthreads();

        // ---- prefetch next X K-block while the matrix pipe works ----
        if (kb + BK < INS) {
            __builtin_prefetch(Xb + (size_t)(kb + BK + (tid >> 3)) * FE + (tid & 7) * 8, 0, 1);
        }

        // ---- compute: 8 K-steps x 4 WMMAs ----
        #pragma unroll
        for (int k0 = 0; k0 < BK; k0 += 4) {
            v2f a0, a1, bv0, bv1;
            {
                const float* p = &Ws[(m_wave +      al) * WS_STRIDE + k0 + ak];
                a0.x = p[0]; a0.y = p[1];
                const float* q = &Ws[(m_wave + 16 + al) * WS_STRIDE + k0 + ak];
                a1.x = q[0]; a1.y = q[1];
            }
            {
                const int pr0 = (k0 + ak) >> 1;   // even K start -> pair row
                bv0 = *(const v2f*)&Xs2[pr0 * XS2_STRIDE + ((n_wave +      al) << 1)];
                bv1 = *(const v2f*)&Xs2[pr0 * XS2_STRIDE + ((n_wave + 16 + al) << 1)];
            }
            c00 = __builtin_amdgcn_wmma_f32_16x16x4_f32(false, a0, false, bv0, (short)0, c00, false, false);
            c01 = __builtin_amdgcn_wmma_f32_16x16x4_f32(false, a0, false, bv1, (short)0, c01, false, false);
            c10 = __builtin_amdgcn_wmma_f32_16x16x4_f32(false, a1, false, bv0, (short)0, c10, false, false);
            c11 = __builtin_amdgcn_wmma_f32_16x16x4_f32(false, a1, false, bv1, (short)0, c11, false, false);
        }
        __syncthreads();
    }

    // ---- writeback: C/D layout = VGPR r: lanes0-15 M=r, lanes16-31 M=r+8 ----
    const int n_out  = al;
    const int m_half = (lane >> 4) << 3;
    #pragma unroll
    for (int r = 0; r < 8; ++r) {
        int m = m_half + r;
        Ob[(size_t)(m_wave +      m) * FE + (n_wave +      n_out)] = c00[r];
        Ob[(size_t)(m_wave +      m) * FE + (n_wave + 16 + n_out)] = c01[r];
        Ob[(size_t)(m_wave + 16 + m) * FE + (n_wave +      n_out)] = c10[r];
        Ob[(size_t)(m_wave + 16 + m) * FE + (n_wave + 16 + n_out)] = c11[r];
    }
}

extern "C" void kernel_launch(void* const* d_in, const int* in_sizes, int n_in,
                              void* d_out, int out_size, void* d_ws, size_t ws_size,
                              hipStream_t stream) {
    (void)in_sizes; (void)n_in; (void)out_size; (void)d_ws; (void)ws_size;
    const float* x   = (const float*)d_in[0];   // [BS, INS, FE]
    const float* wgt = (const float*)d_in[1];   // [BS, OUTS, INS]
    float*       out = (float*)d_out;           // [BS, OUTS, FE]

    dim3 grid(FE / BN, OUTS / BM, BS);          // (32, 8, 32)
    bgemm_wmma_f32<<<grid, dim3(256), 0, stream>>>(x, wgt, out);
}